// Memory_43181601194129
// MI455X (gfx1250) — compile-verified
//
#include <hip/hip_runtime.h>
#include <math.h>

typedef __attribute__((ext_vector_type(2))) float v2f;
typedef __attribute__((ext_vector_type(8))) float v8f;

static constexpr int Bb  = 64;
static constexpr int Dd  = 512;
static constexpr int HW  = 1024;          // 32*32
static constexpr int Mm  = 512;
static constexpr int Nn  = Bb * HW;       // 65536
static constexpr int UQC = 2 * Dd * HW;   // per-batch stride of updated_query (1,048,576)

// ---------------------------------------------------------------------------
// 1) per-token inverse L2 norm over channels
// ---------------------------------------------------------------------------
__global__ __launch_bounds__(256) void k_norm(const float* __restrict__ q,
                                              float* __restrict__ inv) {
  const int b = blockIdx.x;
  const int t = threadIdx.x;
  const float* p = q + (size_t)b * Dd * HW;
  float s0 = 0.f, s1 = 0.f, s2 = 0.f, s3 = 0.f;
  for (int d = 0; d < Dd; ++d) {
    const float* pd = p + (size_t)d * HW;
    float a = pd[t], c = pd[t + 256], e = pd[t + 512], f = pd[t + 768];
    s0 += a * a; s1 += c * c; s2 += e * e; s3 += f * f;
  }
  const int n0 = b * HW;
  inv[n0 + t      ] = 1.f / fmaxf(sqrtf(s0), 1e-12f);
  inv[n0 + t + 256] = 1.f / fmaxf(sqrtf(s1), 1e-12f);
  inv[n0 + t + 512] = 1.f / fmaxf(sqrtf(s2), 1e-12f);
  inv[n0 + t + 768] = 1.f / fmaxf(sqrtf(s3), 1e-12f);
}

// ---------------------------------------------------------------------------
// 2) write normalized query into first half of updated_query [B,2D,H,W]
// ---------------------------------------------------------------------------
__global__ __launch_bounds__(256) void k_writeq(const float* __restrict__ q,
                                                const float* __restrict__ inv,
                                                float* __restrict__ uq) {
  const size_t idx = (size_t)blockIdx.x * 256 + threadIdx.x;   // < B*D*HW
  const int b  = (int)(idx >> 19);         // D*HW = 524288
  const int hw = (int)(idx & 1023);
  const float v = q[idx] * inv[b * HW + hw];
  uq[idx + ((size_t)b << 19)] = v;         // b*2D*HW + d*HW + hw
}

// ---------------------------------------------------------------------------
// 3) GEMM1: score[N,M] = qr[N,D] @ keys[M,D]^T  via V_WMMA_F32_16X16X4_F32
//    qr read from NCHW first half of updated_query (coalesced across lanes)
// ---------------------------------------------------------------------------
__global__ __launch_bounds__(256) void k_score_wmma(const float* __restrict__ uq,
                                                    const float* __restrict__ keys,
                                                    float* __restrict__ score) {
  const int lane = threadIdx.x & 31;
  const int wave = threadIdx.x >> 5;
  const int gid  = blockIdx.x * 8 + wave;
  const int rowBase = (gid >> 3) * 16;     // token tile (16 rows)
  const int colBase = (gid & 7) * 64;      // 4x16 memory-slot columns
  const int half = lane >> 4;
  const int lr   = lane & 15;

  const int b  = rowBase >> 10;            // tile never crosses a batch image
  const int hw = (rowBase & 1023) + lr;
  const float* qbase = uq + (size_t)b * UQC + hw;

  v8f acc0 = {0.f,0.f,0.f,0.f,0.f,0.f,0.f,0.f};
  v8f acc1 = acc0, acc2 = acc0, acc3 = acc0;

  for (int k = 0; k < Dd; k += 4) {
    const int kk = k + 2 * half;
    v2f a;
    a.x = qbase[(size_t)kk * HW];
    a.y = qbase[(size_t)(kk + 1) * HW];
    v2f b0 = *(const v2f*)(keys + (size_t)(colBase +  0 + lr) * Dd + kk);
    v2f b1 = *(const v2f*)(keys + (size_t)(colBase + 16 + lr) * Dd + kk);
    v2f b2 = *(const v2f*)(keys + (size_t)(colBase + 32 + lr) * Dd + kk);
    v2f b3 = *(const v2f*)(keys + (size_t)(colBase + 48 + lr) * Dd + kk);
    acc0 = __builtin_amdgcn_wmma_f32_16x16x4_f32(false, a, false, b0, (short)0, acc0, false, false);
    acc1 = __builtin_amdgcn_wmma_f32_16x16x4_f32(false, a, false, b1, (short)0, acc1, false, false);
    acc2 = __builtin_amdgcn_wmma_f32_16x16x4_f32(false, a, false, b2, (short)0, acc2, false, false);
    acc3 = __builtin_amdgcn_wmma_f32_16x16x4_f32(false, a, false, b3, (short)0, acc3, false, false);
  }
  for (int r = 0; r < 8; ++r) {
    const int row = rowBase + r + 8 * half;
    float* sp = score + (size_t)row * Mm + colBase + lr;
    sp[0]  = acc0[r];
    sp[16] = acc1[r];
    sp[32] = acc2[r];
    sp[48] = acc3[r];
  }
}

// ---------------------------------------------------------------------------
// 4) column softmax statistics (softmax over N) — online max/sum partials
// ---------------------------------------------------------------------------
__global__ __launch_bounds__(512) void k_colpart(const float* __restrict__ score,
                                                 float* __restrict__ pmax,
                                                 float* __restrict__ psum) {
  const int col = threadIdx.x;
  const float* p = score + (size_t)blockIdx.x * 1024 * Mm + col;
  float m = -__builtin_inff(), s = 0.f;
  for (int r = 0; r < 1024; ++r) {
    const float x = p[(size_t)r * Mm];
    const float nm = fmaxf(m, x);
    s = s * __expf(m - nm) + __expf(x - nm);
    m = nm;
  }
  pmax[blockIdx.x * Mm + col] = m;
  psum[blockIdx.x * Mm + col] = s;
}

__global__ __launch_bounds__(512) void k_colfinal(const float* __restrict__ pmax,
                                                  const float* __restrict__ psum,
                                                  float* __restrict__ colmax,
                                                  float* __restrict__ colsum) {
  const int col = threadIdx.x;
  float m = -__builtin_inff();
  for (int k = 0; k < 64; ++k) m = fmaxf(m, pmax[k * Mm + col]);
  float s = 0.f;
  for (int k = 0; k < 64; ++k) s += psum[k * Mm + col] * __expf(pmax[k * Mm + col] - m);
  colmax[col] = m;
  colsum[col] = s;
}

// ---------------------------------------------------------------------------
// 5) per-row pass: softmaxes, top-2, losses, scatter update (one wave per row)
// ---------------------------------------------------------------------------
__device__ __forceinline__ void top2_merge(float v, int i,
                                           float& v1, int& i1, float& v2, int& i2) {
  if (v > v1 || (v == v1 && i < i1)) { v2 = v1; i2 = i1; v1 = v; i1 = i; }
  else if (v > v2 || (v == v2 && i < i2)) { v2 = v; i2 = i; }
}

__global__ __launch_bounds__(256) void k_rowpass(const float* __restrict__ uq,
                                                 const float* __restrict__ keys,
                                                 const float* __restrict__ colmax,
                                                 const float* __restrict__ colsum,
                                                 float* __restrict__ squery,  // in: raw score, out: s_query
                                                 float* __restrict__ smem,    // out: s_memory
                                                 float* __restrict__ qupd,
                                                 float* __restrict__ gloss,
                                                 float* __restrict__ sloss) {
  const int lane = threadIdx.x & 31;
  const int n = blockIdx.x * 8 + (threadIdx.x >> 5);
  float* row = squery + (size_t)n * Mm;

  float x[16];
  float rmax = -__builtin_inff();
  for (int i = 0; i < 16; ++i) { x[i] = row[i * 32 + lane]; rmax = fmaxf(rmax, x[i]); }
  for (int off = 16; off; off >>= 1) rmax = fmaxf(rmax, __shfl_xor(rmax, off, 32));

  float rsum = 0.f;
  for (int i = 0; i < 16; ++i) rsum += __expf(x[i] - rmax);
  for (int off = 16; off; off >>= 1) rsum += __shfl_xor(rsum, off, 32);
  const float rinv = 1.f / rsum;

  // top-2 of the row (ties -> lowest index, matching top_k)
  float v1 = -__builtin_inff(), v2 = -__builtin_inff();
  int i1 = 1 << 30, i2 = 1 << 30;
  for (int i = 0; i < 16; ++i) top2_merge(x[i], i * 32 + lane, v1, i1, v2, i2);
  for (int off = 16; off; off >>= 1) {
    float ov1 = __shfl_xor(v1, off, 32), ov2 = __shfl_xor(v2, off, 32);
    int   oi1 = __shfl_xor(i1, off, 32), oi2 = __shfl_xor(i2, off, 32);
    top2_merge(ov1, oi1, v1, i1, v2, i2);
    top2_merge(ov2, oi2, v1, i1, v2, i2);
  }

  // write softmaxes (s_query rewritten in place over raw score)
  for (int i = 0; i < 16; ++i) {
    const int col = i * 32 + lane;
    smem[(size_t)n * Mm + col] = __expf(x[i] - rmax) * rinv;
    row[col] = __expf(x[i] - colmax[col]) / colsum[col];
  }

  // losses + scatter update.  w_n = s_query[n,g]/max_N s_query[:,g] = exp(rmax - colmax[g])
  const int g1 = i1, g2 = i2;
  const float wn = __expf(rmax - colmax[g1]);
  const int bi = n >> 10, hw = n & 1023;
  const float* qb = uq + (size_t)bi * UQC + hw;
  float gl = 0.f, dp = 0.f, dn = 0.f;
  for (int i = 0; i < 16; ++i) {
    const int d = i * 32 + lane;
    const float qv = qb[(size_t)d * HW];
    const float pv = keys[(size_t)g1 * Dd + d];
    const float nv = keys[(size_t)g2 * Dd + d];
    const float e1 = qv - pv;
    gl += e1 * e1;
    const float a = e1 + 1e-6f;       dp += a * a;
    const float c = qv - nv + 1e-6f;  dn += c * c;
    atomicAdd(&qupd[(size_t)g1 * Dd + d], wn * qv);
  }
  for (int off = 16; off; off >>= 1) {
    gl += __shfl_xor(gl, off, 32);
    dp += __shfl_xor(dp, off, 32);
    dn += __shfl_xor(dn, off, 32);
  }
  if (lane == 0) {
    atomicAdd(&gloss[bi], gl * (1.f / (1024.f * 512.f)));
    const float hinge = fmaxf(sqrtf(dp) - sqrtf(dn) + 1.f, 0.f);
    atomicAdd(&sloss[bi], hinge * (1.f / 1024.f));
  }
}

// ---------------------------------------------------------------------------
// 6) GEMM2: concat_mem[N,D] = s_memory[N,M] @ keys[M,D] -> second half of uq
// ---------------------------------------------------------------------------
__global__ __launch_bounds__(256) void k_read_wmma(const float* __restrict__ sm,
                                                   const float* __restrict__ keys,
                                                   float* __restrict__ uq) {
  const int lane = threadIdx.x & 31;
  const int wave = threadIdx.x >> 5;
  const int gid  = blockIdx.x * 8 + wave;
  const int rowBase = (gid >> 3) * 16;     // tokens
  const int colBase = (gid & 7) * 64;      // output channels d
  const int half = lane >> 4;
  const int lr   = lane & 15;

  v8f acc0 = {0.f,0.f,0.f,0.f,0.f,0.f,0.f,0.f};
  v8f acc1 = acc0, acc2 = acc0, acc3 = acc0;

  const float* arow = sm + (size_t)(rowBase + lr) * Mm;
  for (int k = 0; k < Mm; k += 4) {
    const int kk = k + 2 * half;
    v2f a = *(const v2f*)(arow + kk);
    const float* k0 = keys + (size_t)kk * Dd;
    const float* k1 = keys + (size_t)(kk + 1) * Dd;
    v2f b0, b1, b2, b3;
    b0.x = k0[colBase +  0 + lr]; b0.y = k1[colBase +  0 + lr];
    b1.x = k0[colBase + 16 + lr]; b1.y = k1[colBase + 16 + lr];
    b2.x = k0[colBase + 32 + lr]; b2.y = k1[colBase + 32 + lr];
    b3.x = k0[colBase + 48 + lr]; b3.y = k1[colBase + 48 + lr];
    acc0 = __builtin_amdgcn_wmma_f32_16x16x4_f32(false, a, false, b0, (short)0, acc0, false, false);
    acc1 = __builtin_amdgcn_wmma_f32_16x16x4_f32(false, a, false, b1, (short)0, acc1, false, false);
    acc2 = __builtin_amdgcn_wmma_f32_16x16x4_f32(false, a, false, b2, (short)0, acc2, false, false);
    acc3 = __builtin_amdgcn_wmma_f32_16x16x4_f32(false, a, false, b3, (short)0, acc3, false, false);
  }
  for (int r = 0; r < 8; ++r) {
    const int n = rowBase + r + 8 * half;
    const int bi = n >> 10, hw = n & 1023;
    float* op = uq + (size_t)bi * UQC + (size_t)(Dd + colBase + lr) * HW + hw;
    op[ 0 * HW] = acc0[r];
    op[16 * HW] = acc1[r];
    op[32 * HW] = acc2[r];
    op[48 * HW] = acc3[r];
  }
}

// ---------------------------------------------------------------------------
// 7) updated_memory = l2norm(q_update + keys)
// ---------------------------------------------------------------------------
__global__ __launch_bounds__(256) void k_updmem(const float* __restrict__ qupd,
                                                const float* __restrict__ keys,
                                                float* __restrict__ out) {
  const int m = blockIdx.x, t = threadIdx.x;
  const float v0 = qupd[(size_t)m * Dd + t      ] + keys[(size_t)m * Dd + t      ];
  const float v1 = qupd[(size_t)m * Dd + t + 256] + keys[(size_t)m * Dd + t + 256];
  float ss = v0 * v0 + v1 * v1;
  for (int off = 16; off; off >>= 1) ss += __shfl_xor(ss, off, 32);
  __shared__ float red[8];
  if ((t & 31) == 0) red[t >> 5] = ss;
  __syncthreads();
  float tot = 0.f;
  for (int i = 0; i < 8; ++i) tot += red[i];
  const float inv = 1.f / fmaxf(sqrtf(tot), 1e-12f);
  out[(size_t)m * Dd + t      ] = v0 * inv;
  out[(size_t)m * Dd + t + 256] = v1 * inv;
}

// ---------------------------------------------------------------------------
// zero-init accumulators (harness poisons buffers with 0xAA)
// ---------------------------------------------------------------------------
__global__ void k_zero(float* __restrict__ qupd, float* __restrict__ gl,
                       float* __restrict__ sl) {
  const int i = blockIdx.x * 256 + threadIdx.x;
  if (i < Mm * Dd) qupd[i] = 0.f;
  if (i < Bb) { gl[i] = 0.f; sl[i] = 0.f; }
}

// ---------------------------------------------------------------------------
extern "C" void kernel_launch(void* const* d_in, const int* in_sizes, int n_in,
                              void* d_out, int out_size, void* d_ws, size_t ws_size,
                              hipStream_t stream) {
  const float* query = (const float*)d_in[0];   // [B,D,H,W]
  const float* keys  = (const float*)d_in[1];   // [M,D]

  float* out    = (float*)d_out;
  float* uq     = out;                           // [B,2D,H,W] 67,108,864
  float* umem   = uq + (size_t)Bb * UQC;         // [M,D]        262,144
  float* squery = umem + (size_t)Mm * Dd;        // [N,M]     33,554,432
  float* smem   = squery + (size_t)Nn * Mm;      // [N,M]     33,554,432
  float* gloss  = smem + (size_t)Nn * Mm;        // [B,1]
  float* sloss  = gloss + Bb;                    // [B,1]

  float* ws     = (float*)d_ws;                  // < 2 MB total
  float* inv    = ws;                            // [N]
  float* pmax   = inv + Nn;                      // [64,M]
  float* psum   = pmax + 64 * Mm;                // [64,M]
  float* colmax = psum + 64 * Mm;                // [M]
  float* colsum = colmax + Mm;                   // [M]
  float* qupd   = colsum + Mm;                   // [M,D]

  k_zero      <<<(Mm * Dd + 255) / 256, 256, 0, stream>>>(qupd, gloss, sloss);
  k_norm      <<<Bb, 256, 0, stream>>>(query, inv);
  k_writeq    <<<(Bb * Dd * HW) / 256, 256, 0, stream>>>(query, inv, uq);
  k_score_wmma<<<(Nn / 16) * (Mm / 64) / 8, 256, 0, stream>>>(uq, keys, squery);
  k_colpart   <<<Nn / 1024, 512, 0, stream>>>(squery, pmax, psum);
  k_colfinal  <<<1, 512, 0, stream>>>(pmax, psum, colmax, colsum);
  k_rowpass   <<<Nn / 8, 256, 0, stream>>>(uq, keys, colmax, colsum, squery, smem,
                                           qupd, gloss, sloss);
  k_read_wmma <<<(Nn / 16) * (Dd / 64) / 8, 256, 0, stream>>>(smem, keys, uq);
  k_updmem    <<<Mm, 256, 0, stream>>>(qupd, keys, umem);
}